// RGCN_45801531244854
// MI455X (gfx1250) — compile-verified
//
#include <hip/hip_runtime.h>

typedef float v2f __attribute__((ext_vector_type(2)));
typedef float v8f __attribute__((ext_vector_type(8)));

#define N_NODES_C 50000
#define N_EDGES_C 800000
#define IN_DIM_C 128
#define H_DIM_C 128
#define NUM_RELS_C 8
#define HSTRIDE (NUM_RELS_C * H_DIM_C)   // 1024 floats per node in H buffer
#define LDS_PITCH (IN_DIM_C + 4)         // 132 floats; rows stay 16B-aligned

// ---------------------------------------------------------------------------
// W[r] = sum_b comp[r,b] * V[b], stored in WMMA B-fragment order:
//   Wf float2-index = ((r*8 + ntile)*32 + kstep)*32 + lane
// where n = ntile*16 + (lane&15), k-half = lane>>4, k0 = kstep*4,
// element 0/1 = rows k0+2*(lane>>4), k0+2*(lane>>4)+1 of column n.
// grid: (64, NUM_RELS), block: 256
__global__ void __launch_bounds__(256)
compose_W_frag(const float* __restrict__ comp, const float* __restrict__ V,
               float* __restrict__ Wf) {
    const int r   = blockIdx.y;
    const int idx = blockIdx.x * 256 + threadIdx.x;   // 0..16383  (k*128 + n)
    const int k   = idx >> 7;
    const int n   = idx & 127;
    float acc = 0.0f;
#pragma unroll
    for (int b = 0; b < NUM_RELS_C; ++b)
        acc += comp[r * NUM_RELS_C + b] * V[b * (IN_DIM_C * H_DIM_C) + idx];

    const int ntile = n >> 4;
    const int lm    = n & 15;
    const int kstep = k >> 2;
    const int kr    = k & 3;
    const int lane  = lm + ((kr >> 1) << 4);
    const int elem  = kr & 1;
    Wf[((((r * 8 + ntile) * 32 + kstep) * 32 + lane) << 1) | elem] = acc;
}

// out[n, o] = bias[o]
__global__ void __launch_bounds__(256)
init_bias(const float* __restrict__ bias, float* __restrict__ out) {
    const int i = blockIdx.x * 256 + threadIdx.x;
    if (i < N_NODES_C * H_DIM_C) out[i] = bias[i & (H_DIM_C - 1)];
}

__global__ void __launch_bounds__(256)
relu_inplace(float* __restrict__ x) {
    const int i = blockIdx.x * 256 + threadIdx.x;
    if (i < N_NODES_C * H_DIM_C) x[i] = fmaxf(x[i], 0.0f);
}

// ---------------------------------------------------------------------------
// H[n, r, :] = X[n, :] @ W[r]  via V_WMMA_F32_16X16X4_F32
// grid: (ceil(N_NODES/32), NUM_RELS), block: 256 (8 waves)
// Block tile: M=32 nodes (LDS, clamped), N=128 (wave w owns cols w*16..+15),
// K=128. Each wave keeps two 16x16 accumulators; every B fragment (one
// coalesced global_load_b64 from the swizzled Wf) feeds two WMMAs.
__global__ void __launch_bounds__(256)
rgcn_gemm_wmma(const float* __restrict__ X, const float* __restrict__ Wf,
               float* __restrict__ H) {
    __shared__ float xs[32 * LDS_PITCH];

    const int r   = blockIdx.y;
    const int m0  = blockIdx.x * 32;
    const int tid = threadIdx.x;

    // Cooperative load of 32x128 X tile as float4 (1024 float4, 4 per thread).
    // Row index clamped so the tail block stays in-bounds with EXEC all-ones.
#pragma unroll
    for (int t = 0; t < 4; ++t) {
        const int j    = tid + t * 256;      // float4 index 0..1023
        const int row  = j >> 5;             // /32
        const int col4 = j & 31;
        int rg = m0 + row;
        if (rg >= N_NODES_C) rg = N_NODES_C - 1;
        const float4 v = reinterpret_cast<const float4*>(
                             X + (long)rg * IN_DIM_C)[col4];
        *reinterpret_cast<float4*>(&xs[row * LDS_PITCH + col4 * 4]) = v;
    }
    __syncthreads();

    const int lane = tid & 31;
    const int wave = tid >> 5;
    const int lm   = lane & 15;        // M (for A) / N (for B) index
    const int lh   = lane >> 4;        // K half-select

    const float2* Bp = reinterpret_cast<const float2*>(Wf) +
                       (long)(r * 8 + wave) * 32 * 32 + lane;

    v8f c0 = {0.f, 0.f, 0.f, 0.f, 0.f, 0.f, 0.f, 0.f};
    v8f c1 = {0.f, 0.f, 0.f, 0.f, 0.f, 0.f, 0.f, 0.f};
#pragma unroll 8
    for (int ks = 0; ks < 32; ++ks) {
        const int ka = ks * 4 + lh * 2;
        v2f a0, a1, b;
        a0.x = xs[lm * LDS_PITCH + ka];
        a0.y = xs[lm * LDS_PITCH + ka + 1];
        a1.x = xs[(16 + lm) * LDS_PITCH + ka];
        a1.y = xs[(16 + lm) * LDS_PITCH + ka + 1];
        const float2 bv = Bp[ks * 32];
        b.x = bv.x;
        b.y = bv.y;
        c0 = __builtin_amdgcn_wmma_f32_16x16x4_f32(false, a0, false, b,
                                                   (short)0, c0, false, false);
        c1 = __builtin_amdgcn_wmma_f32_16x16x4_f32(false, a1, false, b,
                                                   (short)0, c1, false, false);
    }

    // D layout: VGPR v -> M = v + 8*lh, N = wave*16 + lm
    float* Ho = H + (long)m0 * HSTRIDE + (long)r * H_DIM_C + wave * 16 + lm;
#pragma unroll
    for (int v = 0; v < 8; ++v) {
        const int m = v + lh * 8;
        Ho[(long)m * HSTRIDE] = c0[v];                       // rows m0..m0+15
        if (m0 + 16 + m < N_NODES_C)
            Ho[(long)(16 + m) * HSTRIDE] = c1[v];            // rows m0+16..+31
    }
}

// ---------------------------------------------------------------------------
// One wave per edge: gather H[src, etype, :], scale by weight, atomic-add to
// agg[dst, :]. Lane handles 4 contiguous dims (float4 load + 4 f32 atomics).
// grid: N_EDGES/8, block: 256 (8 waves)
__global__ void __launch_bounds__(256)
edge_scatter(const float* __restrict__ H, const int* __restrict__ src,
             const int* __restrict__ dst, const int* __restrict__ etypes,
             const float* __restrict__ ew, float* __restrict__ agg,
             int n_edges) {
    const int e    = blockIdx.x * 8 + (threadIdx.x >> 5);
    const int lane = threadIdx.x & 31;
    if (e >= n_edges) return;

    const int   s  = src[e];
    const int   d  = dst[e];
    const int   et = etypes[e];
    const float w  = ew[e];

    const float4 v = reinterpret_cast<const float4*>(
                         H + (long)s * HSTRIDE + (long)et * H_DIM_C)[lane];
    float* ap = agg + (long)d * H_DIM_C + lane * 4;
    __hip_atomic_fetch_add(ap + 0, v.x * w, __ATOMIC_RELAXED, __HIP_MEMORY_SCOPE_AGENT);
    __hip_atomic_fetch_add(ap + 1, v.y * w, __ATOMIC_RELAXED, __HIP_MEMORY_SCOPE_AGENT);
    __hip_atomic_fetch_add(ap + 2, v.z * w, __ATOMIC_RELAXED, __HIP_MEMORY_SCOPE_AGENT);
    __hip_atomic_fetch_add(ap + 3, v.w * w, __ATOMIC_RELAXED, __HIP_MEMORY_SCOPE_AGENT);
}

// ---------------------------------------------------------------------------
extern "C" void kernel_launch(void* const* d_in, const int* in_sizes, int n_in,
                              void* d_out, int out_size, void* d_ws, size_t ws_size,
                              hipStream_t stream) {
    const float* features = (const float*)d_in[0];
    const int*   etypes   = (const int*)  d_in[1];
    const float* eweight  = (const float*)d_in[2];
    const int*   src      = (const int*)  d_in[3];
    const int*   dst      = (const int*)  d_in[4];
    const float* comp1    = (const float*)d_in[5];
    const float* V1       = (const float*)d_in[6];
    const float* bias1    = (const float*)d_in[7];
    const float* comp2    = (const float*)d_in[8];
    const float* V2       = (const float*)d_in[9];
    const float* bias2    = (const float*)d_in[10];
    float* out = (float*)d_out;

    // Workspace layout (bytes):
    //   [0)            W1 (fragment-swizzled): 262144 B
    //   [262144)       W2 (fragment-swizzled): 262144 B
    //   [524288)       H : 50000*1024 f32 = 204800000 B
    //   [205324288)    agg1: 50000*128 f32 = 25600000 B
    char* ws = (char*)d_ws;
    float* W1  = (float*)(ws);
    float* W2  = (float*)(ws + 262144);
    float* H   = (float*)(ws + 524288);
    float* agg = (float*)(ws + 205324288UL);

    const dim3 blk(256);
    const dim3 gW(64, NUM_RELS_C);                      // compose_W_frag
    const dim3 gG((N_NODES_C + 31) / 32, NUM_RELS_C);   // gemm: 1563 x 8
    const int  gE = N_EDGES_C / 8;                      // edge: 100000
    const int  gN = (N_NODES_C * H_DIM_C) / 256;        // elementwise: 25000

    // Relation-composed weights for both layers (fragment layout).
    compose_W_frag<<<gW, blk, 0, stream>>>(comp1, V1, W1);
    compose_W_frag<<<gW, blk, 0, stream>>>(comp2, V2, W2);

    // ---- Layer 1 ----
    init_bias<<<gN, blk, 0, stream>>>(bias1, agg);
    rgcn_gemm_wmma<<<gG, blk, 0, stream>>>(features, W1, H);
    edge_scatter<<<gE, blk, 0, stream>>>(H, src, dst, etypes, eweight, agg, N_EDGES_C);
    relu_inplace<<<gN, blk, 0, stream>>>(agg);

    // ---- Layer 2 ----
    init_bias<<<gN, blk, 0, stream>>>(bias2, out);
    rgcn_gemm_wmma<<<gG, blk, 0, stream>>>(agg, W2, H);
    edge_scatter<<<gE, blk, 0, stream>>>(H, src, dst, etypes, eweight, out, N_EDGES_C);
}